// LightRDL_38706245272171
// MI455X (gfx1250) — compile-verified
//
#include <hip/hip_runtime.h>
#include <hip/hip_bf16.h>

typedef __attribute__((ext_vector_type(2))) float v2f;
typedef __attribute__((ext_vector_type(8))) float v8f;

#define GEMM_ACC  1
#define GEMM_BIAS 2
#define GEMM_RELU 4

// ---------------------------------------------------------------------------
// fp32 WMMA GEMM:  C[M x Nout] (+)= A[M x K] * B[K x Nout] (+bias)(relu)
// Block = 128 threads = 4 waves arranged 2x2; block tile = 32 rows x 32 cols.
// A tile (32r x 32k) and B tile (32k x 32c, stored transposed) staged in LDS,
// zero-padded past M/K/Nout. Full-tile fast paths stage without guards
// (b128 loads when lda allows); inner loop is branch-free and fully unrolled:
// 8 x v_wmma_f32_16x16x4_f32 per chunk, operands via ds_load_2addr_b64.
// EXEC is all-ones through every WMMA (guards only at the final store).
// ---------------------------------------------------------------------------
__global__ __launch_bounds__(128)
void gemm_wmma_f32(const float* __restrict__ A, int lda,
                   const float* __restrict__ B, int ldb,
                   const float* __restrict__ bias,
                   float* __restrict__ C, int ldc,
                   int M, int K, int Nout, int flags)
{
    __shared__ float As[32][36];   // [row][k]  row stride 144B (16B aligned)
    __shared__ float Bs[32][34];   // [col][k]  row stride 136B (8B aligned)

    const int tid   = threadIdx.x;
    const int lane  = tid & 31;
    const int wave  = tid >> 5;
    const int row0  = blockIdx.x << 5;            // 32 rows per block
    const int colB  = blockIdx.y << 5;            // 32 cols per block
    const int wrow  = (wave >> 1) << 4;           // wave's 16-row sub-tile
    const int wcol  = (wave & 1) << 4;            // wave's 16-col sub-tile
    const int mrow  = lane & 15;
    const int khalf = lane >> 4;                  // lanes 0-15: k,k+1 ; 16-31: k+2,k+3

    const bool fullM = (row0 + 32 <= M);
    const bool fullN = (colB + 32 <= Nout);
    const bool vecA  = ((lda & 3) == 0);

    v8f acc = {};
    for (int k0 = 0; k0 < K; k0 += 32) {
        const bool fullK = (k0 + 32 <= K);

        // ---- stage A (32 rows x 32 k) ----
        if (fullM && fullK) {
            if (vecA) {
                // 256 float4 slots, 2 per thread: global_load_b128 + ds_store_b128
                #pragma unroll
                for (int i = 0; i < 2; ++i) {
                    int id = tid + i * 128;
                    int r  = id >> 3;            // 8 float4 per row
                    int q  = (id & 7) << 2;
                    float4 v = *(const float4*)&A[(size_t)(row0 + r) * lda + k0 + q];
                    *(float4*)&As[r][q] = v;
                }
            } else {
                #pragma unroll
                for (int i = 0; i < 8; ++i) {
                    int id = tid + i * 128;
                    int r  = id >> 5;
                    int kk = id & 31;
                    As[r][kk] = A[(size_t)(row0 + r) * lda + k0 + kk];
                }
            }
        } else {
            #pragma unroll
            for (int i = 0; i < 8; ++i) {
                int id  = tid + i * 128;
                int r   = id >> 5;
                int kk  = id & 31;
                int row = row0 + r;
                int kg  = k0 + kk;
                As[r][kk] = (row < M && kg < K) ? A[(size_t)row * lda + kg] : 0.0f;
            }
        }

        // ---- stage B transposed: Bs[col][k] ----
        if (fullN && fullK) {
            #pragma unroll
            for (int i = 0; i < 8; ++i) {
                int id = tid + i * 128;
                int kk = id >> 5;
                int c  = id & 31;
                Bs[c][kk] = B[(size_t)(k0 + kk) * ldb + colB + c];
            }
        } else {
            #pragma unroll
            for (int i = 0; i < 8; ++i) {
                int id  = tid + i * 128;
                int kk  = id >> 5;
                int c   = id & 31;
                int kg  = k0 + kk;
                int col = colB + c;
                Bs[c][kk] = (kg < K && col < Nout) ? B[(size_t)kg * ldb + col] : 0.0f;
            }
        }

        // prefetch next A chunk (one cacheline touch per row)
        if (tid < 32 && k0 + 32 < K) {
            int row = row0 + tid;
            if (row < M)
                __builtin_prefetch(&A[(size_t)row * lda + k0 + 32], 0, 1);
        }
        __syncthreads();

        // ---- 8 chained WMMAs, branch-free ----
        #pragma unroll
        for (int kk = 0; kk < 32; kk += 4) {
            v2f a, b;
            a.x = As[wrow + mrow][kk + 2 * khalf];
            a.y = As[wrow + mrow][kk + 2 * khalf + 1];
            b.x = Bs[wcol + mrow][kk + 2 * khalf];
            b.y = Bs[wcol + mrow][kk + 2 * khalf + 1];
            acc = __builtin_amdgcn_wmma_f32_16x16x4_f32(
                    false, a, false, b, (short)0, acc, false, false);
        }
        __syncthreads();
    }

    // ---- epilogue ----
    const int col = colB + wcol + mrow;
    const int colSafe = (col < Nout) ? col : (Nout - 1);
    float bv = 0.0f;
    if (flags & GEMM_BIAS) bv = bias[colSafe];
    #pragma unroll
    for (int r = 0; r < 8; ++r) {
        int row = row0 + wrow + r + 8 * khalf;    // C tile layout (ISA 7.12.2)
        if (col < Nout && row < M) {
            size_t idx = (size_t)row * ldc + col;
            float v = acc[r] + bv;
            if (flags & GEMM_ACC)  v += C[idx];
            if (flags & GEMM_RELU) v = fmaxf(v, 0.0f);
            C[idx] = v;
        }
    }
}

// ---------------------------------------------------------------------------
__global__ void fill_f32(float* __restrict__ p, long long n, float v) {
    long long t = (long long)blockIdx.x * blockDim.x + threadIdx.x;
    if (t < n) p[t] = v;
}

// segment-sum over edges: agg[dst] += xsrc[src], cnt[dst] += 1
__global__ void scatter_add_kernel(const float* __restrict__ xsrc, int width,
                                   const int* __restrict__ srcIdx,
                                   const int* __restrict__ dstIdx,
                                   long long E, float* __restrict__ agg,
                                   float* __restrict__ cnt)
{
    long long t = (long long)blockIdx.x * blockDim.x + threadIdx.x;
    long long total = E * (long long)width;
    if (t >= total) return;
    int e = (int)(t / width);
    int f = (int)(t % width);
    int s = srcIdx[e];
    int d = dstIdx[e];
    __hip_atomic_fetch_add(&agg[(size_t)d * width + f],
                           xsrc[(size_t)s * width + f],
                           __ATOMIC_RELAXED, __HIP_MEMORY_SCOPE_AGENT);
    if (f == 0)
        __hip_atomic_fetch_add(&cnt[d], 1.0f,
                               __ATOMIC_RELAXED, __HIP_MEMORY_SCOPE_AGENT);
}

__global__ void mean_div(float* __restrict__ agg, const float* __restrict__ cnt,
                         long long rows, int width)
{
    long long t = (long long)blockIdx.x * blockDim.x + threadIdx.x;
    if (t >= rows * (long long)width) return;
    long long r = t / width;
    float c = cnt[r];
    c = (c > 1.0f) ? c : 1.0f;
    agg[t] = agg[t] / c;
}

// x_dr[:,10:20] = x_drivers[:,1500:1510]
__global__ void copy_tail10(const float* __restrict__ xdrv, float* __restrict__ xdr,
                            long long N)
{
    long long t = (long long)blockIdx.x * blockDim.x + threadIdx.x;
    if (t >= N * 10) return;
    long long r = t / 10;
    int j = (int)(t % 10);
    xdr[r * 20 + 10 + j] = xdrv[(size_t)r * 1510 + 1500 + j];
}

// out = leaky_relu(nd + residual), residual = xprev[:, -10:] into cols 246..255
__global__ void act_leaky(const float* __restrict__ nd, const float* __restrict__ xprev,
                          int prevW, int hasRes, float* __restrict__ out, long long rows)
{
    long long t = (long long)blockIdx.x * blockDim.x + threadIdx.x;
    if (t >= rows * 256) return;
    long long r = t >> 8;
    int c = (int)(t & 255);
    float v = nd[t];
    if (hasRes && c >= 246)
        v += xprev[(size_t)r * prevW + (prevW - 10) + (c - 246)];
    out[t] = (v > 0.0f) ? v : 0.01f * v;
}

// ---------------------------------------------------------------------------
extern "C" void kernel_launch(void* const* d_in, const int* in_sizes, int n_in,
                              void* d_out, int out_size, void* d_ws, size_t ws_size,
                              hipStream_t stream)
{
    (void)out_size; (void)ws_size;
    auto F  = [&](int i) { return (const float*)d_in[i]; };
    auto II = [&](int i) { return (const int*)d_in[i]; };

    const float *xdrv, *xra_in;
    const int *eDR2RA, *eRA2DR, *eDR2DR;
    const float *W1, *b1, *W2, *b2;
    const float *Wl[2][3], *bl[2][3], *Wr[2][3];  // [layer][0=dr2ra,1=ra2dr,2=dr2dr]
    const float *lin0W, *lin0b, *linW, *linb;

    // Variant A: insertion order (x_drivers first). Variant B: jax-sorted pytree.
    bool varB = (in_sizes[0] < 5000000);
    long long N, E;
    if (!varB) {
        xdrv = F(0); xra_in = F(1);
        eDR2RA = II(2); eRA2DR = II(3); eDR2DR = II(4);
        W1 = F(5); b1 = F(6); W2 = F(7); b2 = F(8);
        int k = 9;
        for (int l = 0; l < 2; ++l)
            for (int rr = 0; rr < 3; ++rr) {        // dr2ra, ra2dr, dr2dr
                Wl[l][rr] = F(k++); bl[l][rr] = F(k++); Wr[l][rr] = F(k++);
            }
        lin0W = F(27); lin0b = F(28); linW = F(29); linb = F(30);
        N = in_sizes[0] / 1510; E = in_sizes[2] / 2;
    } else {
        eDR2DR = II(0); eDR2RA = II(1); eRA2DR = II(2);
        int k = 3;
        const int order[3] = {2, 0, 1};             // sorted: dr2dr, dr2ra, ra2dr
        for (int l = 0; l < 2; ++l)
            for (int rs = 0; rs < 3; ++rs) {
                int rr = order[rs];
                Wl[l][rr] = F(k++); Wr[l][rr] = F(k++); bl[l][rr] = F(k++);
            }
        linW = F(21); linb = F(22); lin0W = F(23); lin0b = F(24);
        W1 = F(25); W2 = F(26); b1 = F(27); b2 = F(28);
        xdrv = F(n_in - 2); xra_in = F(n_in - 1);
        N = in_sizes[n_in - 2] / 1510; E = in_sizes[0] / 2;
    }

    // workspace layout (floats)
    float* ws = (float*)d_ws;
    size_t nb = (size_t)N * 256;
    float* bufA  = ws;            // x_dr layer0 (w20, ld20); reused as final x_dr
    float* bufB  = ws + 1 * nb;   // new_dr accumulator
    float* bufC  = ws + 2 * nb;   // x_dr layer1 input (w256)
    float* bufRA = ws + 3 * nb;   // x_ra layer1 input (w256)
    float* bufNR = ws + 4 * nb;   // new_ra accumulator / head hidden
    float* agg   = ws + 5 * nb;   // scatter accumulator
    float* t1    = ws + 6 * nb;   // MLP hidden (N x 20)
    float* cnt   = ws + 6 * nb + (size_t)N * 20;

    auto fill = [&](float* p, long long n) {
        fill_f32<<<(unsigned)((n + 255) / 256), 256, 0, stream>>>(p, n, 0.0f);
    };
    auto gemm = [&](const float* A, int lda, const float* B, int ldb,
                    const float* bias, float* C, int ldc, int K, int Nout, int flags) {
        dim3 grid((unsigned)((N + 31) / 32), (unsigned)((Nout + 31) / 32));
        gemm_wmma_f32<<<grid, 128, 0, stream>>>(A, lda, B, ldb, bias, C, ldc,
                                                (int)N, K, Nout, flags);
    };
    auto scat = [&](const float* xs, int w, const int* ei) {
        long long tot = E * (long long)w;
        scatter_add_kernel<<<(unsigned)((tot + 255) / 256), 256, 0, stream>>>(
            xs, w, ei, ei + E, E, agg, cnt);
    };
    auto mean = [&](int w) {
        long long tot = N * (long long)w;
        mean_div<<<(unsigned)((tot + 255) / 256), 256, 0, stream>>>(agg, cnt, N, w);
    };

    // ---- front MLP: x_dr = [relu(x[:,:1500]@W1+b1)@W2+b2 , x[:,1500:]] ----
    gemm(xdrv, 1510, W1, 20, b1, t1, 20, 1500, 20, GEMM_BIAS | GEMM_RELU);
    gemm(t1, 20, W2, 10, b2, bufA, 20, 20, 10, GEMM_BIAS);
    copy_tail10<<<(unsigned)((N * 10 + 255) / 256), 256, 0, stream>>>(xdrv, bufA, N);

    // ---- layer 0 ----
    // dr2ra -> new_ra (bufNR)
    fill(agg, N * 20); fill(cnt, N);
    scat(bufA, 20, eDR2RA); mean(20);
    gemm(agg, 20, Wl[0][0], 256, bl[0][0], bufNR, 256, 20, 256, GEMM_BIAS);
    gemm(xra_in, 128, Wr[0][0], 256, nullptr, bufNR, 256, 128, 256, GEMM_ACC);
    // ra2dr -> new_dr (bufB)
    fill(agg, N * 128); fill(cnt, N);
    scat(xra_in, 128, eRA2DR); mean(128);
    gemm(agg, 128, Wl[0][1], 256, bl[0][1], bufB, 256, 128, 256, GEMM_BIAS);
    gemm(bufA, 20, Wr[0][1], 256, nullptr, bufB, 256, 20, 256, GEMM_ACC);
    // dr2dr -> new_dr (accumulate)
    fill(agg, N * 20); fill(cnt, N);
    scat(bufA, 20, eDR2DR); mean(20);
    gemm(agg, 20, Wl[0][2], 256, bl[0][2], bufB, 256, 20, 256, GEMM_ACC | GEMM_BIAS);
    gemm(bufA, 20, Wr[0][2], 256, nullptr, bufB, 256, 20, 256, GEMM_ACC);
    // activations
    act_leaky<<<(unsigned)((N * 256 + 255) / 256), 256, 0, stream>>>(bufB, bufA, 20, 1, bufC, N);
    act_leaky<<<(unsigned)((N * 256 + 255) / 256), 256, 0, stream>>>(bufNR, bufNR, 0, 0, bufRA, N);

    // ---- layer 1 (dr2ra branch is dead: head only consumes x_dr) ----
    // ra2dr -> new_dr (bufB)
    fill(agg, N * 256); fill(cnt, N);
    scat(bufRA, 256, eRA2DR); mean(256);
    gemm(agg, 256, Wl[1][1], 256, bl[1][1], bufB, 256, 256, 256, GEMM_BIAS);
    gemm(bufC, 256, Wr[1][1], 256, nullptr, bufB, 256, 256, 256, GEMM_ACC);
    // dr2dr -> new_dr (accumulate)
    fill(agg, N * 256); fill(cnt, N);
    scat(bufC, 256, eDR2DR); mean(256);
    gemm(agg, 256, Wl[1][2], 256, bl[1][2], bufB, 256, 256, 256, GEMM_ACC | GEMM_BIAS);
    gemm(bufC, 256, Wr[1][2], 256, nullptr, bufB, 256, 256, 256, GEMM_ACC);
    // activation with residual from bufC (width 256) -> bufA
    act_leaky<<<(unsigned)((N * 256 + 255) / 256), 256, 0, stream>>>(bufB, bufC, 256, 1, bufA, N);

    // ---- head: out = relu(x_dr@W0+b0)@Wf+bf ----
    gemm(bufA, 256, lin0W, 128, lin0b, bufNR, 128, 256, 128, GEMM_BIAS | GEMM_RELU);
    gemm(bufNR, 128, linW, 2, linb, (float*)d_out, 2, 128, 2, GEMM_BIAS);
}